// Memory_2774548873902
// MI455X (gfx1250) — compile-verified
//
#include <hip/hip_runtime.h>
#include <hip/hip_bf16.h>
#include <math.h>

typedef float v2f __attribute__((ext_vector_type(2)));
typedef float v8f __attribute__((ext_vector_type(8)));

#define M_TILE 16
#define NMEM   2000
#define NMEMP  2004   /* padded row stride: 2004 % 64 == 20 -> conflict-free GEMM2 A reads */
#define CDIM   256
#define NT     125    /* NMEM / 16 */
#define DELTA  0.00025f
#define EPSF   1e-14f

// ---------------------------------------------------------------------------
// Kernel 1: per-row L2 norms of mem (2000 x 256) -> d_ws. One wave per row.
// ---------------------------------------------------------------------------
__global__ __launch_bounds__(256) void mem_norm_kernel(const float* __restrict__ mem,
                                                       float* __restrict__ mnorm) {
  const int lane = threadIdx.x & 31;
  const int row  = blockIdx.x * (blockDim.x >> 5) + (threadIdx.x >> 5);
  if (row >= NMEM) return;
  const float* r = mem + (size_t)row * CDIM;
  float s = 0.f;
  for (int c = lane; c < CDIM; c += 32) { float v = r[c]; s += v * v; }
  #pragma unroll
  for (int off = 16; off > 0; off >>= 1) s += __shfl_xor(s, off, 32);
  if (lane == 0) mnorm[row] = sqrtf(s);
}

// ---------------------------------------------------------------------------
// Kernel 2: fused  dot -> cos -> softmax -> shrink -> renorm -> z_hat
// One block = 8 waves handles a 16-row M tile. All intermediates live in LDS.
// ---------------------------------------------------------------------------
__global__ __launch_bounds__(256) void memory_fused_kernel(const float* __restrict__ z,
                                                           const float* __restrict__ mem,
                                                           const float* __restrict__ mnorm,
                                                           float* __restrict__ out_zhat,
                                                           float* __restrict__ out_what) {
  __shared__ float s_row[M_TILE][NMEMP];     // 128,256 B : dot / cos / exp / w_hat, in place
  __shared__ float s_z[M_TILE][CDIM + 4];    //  16,640 B : z tile (padded)
  __shared__ float s_mn[NMEM];               //   8,000 B : mem norms
  __shared__ float s_znorm[M_TILE];

  const int tid    = threadIdx.x;
  const int lane   = tid & 31;
  const int wave   = tid >> 5;     // 0..7
  const int half   = lane >> 4;    // 0 or 1  (lanes 16-31 carry K+2..K+3)
  const int l16    = lane & 15;
  const int m_base = blockIdx.x * M_TILE;

  // ---- Phase 0: stage z tile (coalesced float4) and mem norms into LDS ----
  {
    const float4* zg = (const float4*)(z + (size_t)m_base * CDIM);
    for (int i = tid; i < M_TILE * CDIM / 4; i += 256) {
      float4 v = zg[i];
      int r = (i * 4) / CDIM, c = (i * 4) % CDIM;
      s_z[r][c] = v.x; s_z[r][c + 1] = v.y; s_z[r][c + 2] = v.z; s_z[r][c + 3] = v.w;
    }
    for (int i = tid; i < NMEM; i += 256) s_mn[i] = mnorm[i];
  }
  __syncthreads();

  // ---- z-row norms: wave w owns rows w and w+8 ----
  for (int r = wave; r < M_TILE; r += 8) {
    float s = 0.f;
    for (int c = lane; c < CDIM; c += 32) { float v = s_z[r][c]; s += v * v; }
    #pragma unroll
    for (int off = 16; off > 0; off >>= 1) s += __shfl_xor(s, off, 32);
    if (lane == 0) s_znorm[r] = sqrtf(s);
  }
  __syncthreads();

  // ---- GEMM1: dot[16 x 2000] = Z_tile(16x256) * MEM^T, f32 WMMA 16x16x4 ----
  for (int nt = wave; nt < NT; nt += 8) {            // wave-uniform -> EXEC all 1s
    const int n = nt * 16 + l16;
    const float* bptr = mem + (size_t)n * CDIM;      // B(k,n) = mem[n][k]
    v8f acc = {};
    for (int k0 = 0; k0 < CDIM; k0 += 4) {
      const int kk = k0 + 2 * half;
      v2f a; a.x = s_z[l16][kk];  a.y = s_z[l16][kk + 1];   // A(m=l16, K=kk..kk+1)
      v2f b; b.x = bptr[kk];      b.y = bptr[kk + 1];        // B(K, n)
      acc = __builtin_amdgcn_wmma_f32_16x16x4_f32(false, a, false, b,
                                                  (short)0, acc, false, false);
    }
    #pragma unroll
    for (int j = 0; j < 8; ++j)                       // D: M = j + 8*half, N = l16
      s_row[j + 8 * half][nt * 16 + l16] = acc[j];
  }
  __syncthreads();

  // ---- softmax + hard shrinkage + L1 renorm, wave-per-row, in LDS ----
  for (int r = wave; r < M_TILE; r += 8) {
    const float zn = s_znorm[r];
    float mx = -3.0e38f;
    for (int n = lane; n < NMEM; n += 32) {
      float c = s_row[r][n] / (zn * s_mn[n] + EPSF);
      s_row[r][n] = c;
      mx = fmaxf(mx, c);
    }
    #pragma unroll
    for (int off = 16; off > 0; off >>= 1) mx = fmaxf(mx, __shfl_xor(mx, off, 32));

    float sum = 0.f;
    for (int n = lane; n < NMEM; n += 32) {
      float e = expf(s_row[r][n] - mx);
      s_row[r][n] = e;
      sum += e;
    }
    #pragma unroll
    for (int off = 16; off > 0; off >>= 1) sum += __shfl_xor(sum, off, 32);
    const float isum = 1.0f / sum;

    float l1 = 0.f;
    for (int n = lane; n < NMEM; n += 32) {
      float w   = s_row[r][n] * isum;
      float tmp = w - DELTA;
      float wh  = (fmaxf(tmp, 0.f) / (fabsf(tmp) + EPSF)) * w;
      s_row[r][n] = wh;
      l1 += wh;                                   // wh >= 0
    }
    #pragma unroll
    for (int off = 16; off > 0; off >>= 1) l1 += __shfl_xor(l1, off, 32);
    const float rinv = 1.0f / (l1 + EPSF);

    float* wout = out_what + (size_t)(m_base + r) * NMEM;
    for (int n = lane; n < NMEM; n += 32) {
      float v = s_row[r][n] * rinv;
      s_row[r][n] = v;
      wout[n] = v;                                // coalesced w_hat store
    }
  }
  __syncthreads();

  // ---- GEMM2: z_hat[16 x 256] = w_hat(16x2000, LDS) * MEM(2000x256) ----
  for (int ct = wave; ct < CDIM / 16; ct += 8) {   // 2 column tiles per wave
    const int cb = ct * 16 + l16;
    v8f acc = {};
    for (int n0 = 0; n0 < NMEM; n0 += 4) {
      const int nk = n0 + 2 * half;
      v2f a; a.x = s_row[l16][nk];                 // A(m=l16, K=nk..nk+1)
      a.y = s_row[l16][nk + 1];
      v2f b; b.x = mem[(size_t)nk * CDIM + cb];    // B(K=nk, n=cb) coalesced over l16
      b.y = mem[(size_t)(nk + 1) * CDIM + cb];
      acc = __builtin_amdgcn_wmma_f32_16x16x4_f32(false, a, false, b,
                                                  (short)0, acc, false, false);
    }
    #pragma unroll
    for (int j = 0; j < 8; ++j)
      out_zhat[(size_t)(m_base + j + 8 * half) * CDIM + ct * 16 + l16] = acc[j];
  }
}

// ---------------------------------------------------------------------------
extern "C" void kernel_launch(void* const* d_in, const int* in_sizes, int n_in,
                              void* d_out, int out_size, void* d_ws, size_t ws_size,
                              hipStream_t stream) {
  const float* z   = (const float*)d_in[0];   // (8,2,32,32,256) -> (16384, 256)
  const float* mem = (const float*)d_in[1];   // (2000, 256)
  float* out_zhat  = (float*)d_out;                        // 16384*256
  float* out_what  = out_zhat + (size_t)16384 * CDIM;      // 16384*2000
  float* mnorm     = (float*)d_ws;                         // 2000 floats

  mem_norm_kernel<<<(NMEM + 7) / 8, 256, 0, stream>>>(mem, mnorm);
  memory_fused_kernel<<<16384 / M_TILE, 256, 0, stream>>>(z, mem, mnorm,
                                                          out_zhat, out_what);
}